// GGNN_LCG_14104672600352
// MI455X (gfx1250) — compile-verified
//
#include <hip/hip_runtime.h>
#include <math.h>

// ---------------------------------------------------------------------------
// f32 WMMA GEMM:  C[M,N] = act( A[M,K] @ W[N,K]^T + bias[N] )
//   MODE 0: A row m, col k  = A0[m*K + k]
//   MODE 1: (K==256) A row m: cols 0..127 from A0[m], cols 128..255 from A1[m^1]
//           (fused concat + literal/negated-literal swap for the l-GRU input)
// Block: 256 threads (8 waves). Wave w -> rows [bx*128 + w*16, +16),
// cols [by*128, +128) as 8 WMMA 16x16 tiles. K fully unrolled in steps of 4
// using V_WMMA_F32_16X16X4_F32. Per k-step: 1 global_load_b64 (A frag, imm
// offset) + 8 ds_load_b64 (B frags, 16-bit imm offsets) issued together, then
// 8 WMMAs -- loads stay in flight, dscnt waits cascade instead of blocking.
// W chunk staged in padded LDS (64-bank conflict-free b64 reads). A-row is
// clamped (not branched) so the inner loop has no EXEC divergence.
// ---------------------------------------------------------------------------

typedef float v2f __attribute__((ext_vector_type(2)));
typedef float v8f __attribute__((ext_vector_type(8)));

template <int MODE, int KV, bool RELU>
__global__ void __launch_bounds__(256)
gemm_wmma_kernel(const float* __restrict__ A0,
                 const float* __restrict__ A1,
                 const float* __restrict__ W,
                 const float* __restrict__ bias,
                 float* __restrict__ C,
                 int M, int N) {
    constexpr int STRIDE = KV + 4;          // +4 floats pad: conflict-free b64 reads
    extern __shared__ float wlds[];         // [128][STRIDE]

    const int tid = threadIdx.x;
    const int n0  = blockIdx.y * 128;

    // ---- stage W[n0..n0+127][0..KV) into LDS (coalesced float4 loads) ----
    const float* wsrc = W + (size_t)n0 * KV;
    constexpr int ROWQ = KV / 4;
    for (int idx = tid; idx < 128 * ROWQ; idx += 256) {
        const int r = idx / ROWQ, c4 = idx % ROWQ;
        const float4 v = ((const float4*)(wsrc + (size_t)r * KV))[c4];
        *(float4*)(wlds + r * STRIDE + c4 * 4) = v;   // 16B-aligned
    }
    __syncthreads();

    const int wave = tid >> 5;
    const int lane = tid & 31;
    const int half = lane >> 4;             // 0: K=k0,k0+1   1: K=k0+2,k0+3
    const int l16  = lane & 15;
    const int m0   = blockIdx.x * 128 + wave * 16;

    // branch-free row clamp: out-of-range rows compute garbage that is never
    // stored (store path is row-guarded); keeps EXEC uniform for WMMA.
    int row = m0 + l16;
    if (row >= M) row = M - 1;

    const float* apA;
    const float* apB = nullptr;
    if (MODE == 0) {
        apA = A0 + (size_t)row * KV + 2 * half;
    } else {
        apA = A0 + (size_t)row * 128 + 2 * half;
        apB = A1 + (size_t)(row ^ 1) * 128 + 2 * half;
    }
    const float* bbase = wlds + l16 * STRIDE + 2 * half;   // 8B-aligned

    v8f acc[8] = {};                        // 8 tiles x 8 VGPRs f32 accumulators

#pragma unroll
    for (int k0 = 0; k0 < KV; k0 += 4) {
        // compile-time source select after unroll (MODE 1 concat boundary)
        const float* ap = (MODE == 0 || k0 < 128) ? (apA + k0) : (apB + (k0 - 128));
        const v2f a = *(const v2f*)ap;      // global_load_b64, imm offset

        v2f b[8];
#pragma unroll
        for (int t = 0; t < 8; ++t)         // 8 ds_load_b64 in flight
            b[t] = *(const v2f*)(bbase + t * 16 * STRIDE + k0);

#pragma unroll
        for (int t = 0; t < 8; ++t)
            acc[t] = __builtin_amdgcn_wmma_f32_16x16x4_f32(
                false, a, false, b[t], (short)0, acc[t], false, false);
    }

    // ---- epilogue: bias (+ReLU) and store. D layout: VGPR v -> row m0+v+8*half,
    //      col n0 + tile*16 + (lane&15) ----
    if (m0 + 16 <= M) {                     // uniform fast path (always taken here)
#pragma unroll
        for (int t = 0; t < 8; ++t) {
            const int col = n0 + t * 16 + l16;
            const float bv = bias[col];
            float* cp = C + (size_t)(m0 + 8 * half) * N + col;
#pragma unroll
            for (int v = 0; v < 8; ++v) {
                float val = acc[t][v] + bv;
                if (RELU) val = fmaxf(val, 0.0f);
                cp[(size_t)v * N] = val;
            }
        }
    } else {
#pragma unroll
        for (int t = 0; t < 8; ++t) {
            const int col = n0 + t * 16 + l16;
            const float bv = bias[col];
#pragma unroll
            for (int v = 0; v < 8; ++v) {
                const int r = m0 + v + 8 * half;
                if (r < M) {
                    float val = acc[t][v] + bv;
                    if (RELU) val = fmaxf(val, 0.0f);
                    C[(size_t)r * N + col] = val;
                }
            }
        }
    }
}

// ---------------------------------------------------------------------------
// segment-sum scatter: one wave per edge, float4 gather, hw f32 atomics
// ---------------------------------------------------------------------------
__global__ void scatter_add_kernel(const float* __restrict__ feat,
                                   const int* __restrict__ gather_idx,
                                   const int* __restrict__ scatter_idx,
                                   float* __restrict__ out, int nE) {
    const int e = blockIdx.x * 8 + (threadIdx.x >> 5);
    if (e >= nE) return;
    const int lane = threadIdx.x & 31;
    const int s = gather_idx[e];
    const int d = scatter_idx[e];
    const float4 v = ((const float4*)(feat + (size_t)s * 128))[lane];
    float* op = out + (size_t)d * 128 + (size_t)lane * 4;
    unsafeAtomicAdd(op + 0, v.x);
    unsafeAtomicAdd(op + 1, v.y);
    unsafeAtomicAdd(op + 2, v.z);
    unsafeAtomicAdd(op + 3, v.w);
}

// ---------------------------------------------------------------------------
// GRU gate fusion: h' = (1-z)*n + z*h, gates from gi/gh [M,384] (r,z,n order)
// ---------------------------------------------------------------------------
__global__ void gru_gate_kernel(const float* __restrict__ gi,
                                const float* __restrict__ gh,
                                const float* __restrict__ h,
                                float* __restrict__ hn, int M) {
    const size_t i = (size_t)blockIdx.x * 256 + threadIdx.x;
    if (i >= (size_t)M * 128) return;
    const int row = (int)(i >> 7), col = (int)(i & 127);
    const float* gir = gi + (size_t)row * 384;
    const float* ghr = gh + (size_t)row * 384;
    const float r = 1.0f / (1.0f + expf(-(gir[col]       + ghr[col])));
    const float z = 1.0f / (1.0f + expf(-(gir[col + 128] + ghr[col + 128])));
    const float n = tanhf(gir[col + 256] + r * ghr[col + 256]);
    hn[i] = (1.0f - z) * n + z * h[i];
}

// ---------------------------------------------------------------------------
static void launch_gemm(int mode, int Kv, bool relu,
                        const float* A0, const float* A1,
                        const float* W, const float* bias, float* C,
                        int M, int N, hipStream_t s) {
    dim3 grid((M + 127) / 128, N / 128), block(256);
    const size_t lds = (size_t)128 * (Kv + 4) * sizeof(float);
    if (mode == 1) {
        hipLaunchKernelGGL((gemm_wmma_kernel<1, 256, false>), grid, block, lds, s,
                           A0, A1, W, bias, C, M, N);
    } else if (relu) {
        hipLaunchKernelGGL((gemm_wmma_kernel<0, 128, true>), grid, block, lds, s,
                           A0, A1, W, bias, C, M, N);
    } else {
        hipLaunchKernelGGL((gemm_wmma_kernel<0, 128, false>), grid, block, lds, s,
                           A0, A1, W, bias, C, M, N);
    }
}

extern "C" void kernel_launch(void* const* d_in, const int* in_sizes, int n_in,
                              void* d_out, int out_size, void* d_ws, size_t ws_size,
                              hipStream_t stream) {
    (void)n_in; (void)out_size; (void)ws_size;

    const int*   l_edge = (const int*)d_in[2];
    const int*   c_edge = (const int*)d_in[3];
    const float* l_emb0 = (const float*)d_in[4];
    const float* c_emb0 = (const float*)d_in[5];
    const float* l2c_w1 = (const float*)d_in[6],  *l2c_b1 = (const float*)d_in[7];
    const float* l2c_w2 = (const float*)d_in[8],  *l2c_b2 = (const float*)d_in[9];
    const float* c2l_w1 = (const float*)d_in[10], *c2l_b1 = (const float*)d_in[11];
    const float* c2l_w2 = (const float*)d_in[12], *c2l_b2 = (const float*)d_in[13];
    const float* c_w_ih = (const float*)d_in[14], *c_w_hh = (const float*)d_in[15];
    const float* c_b_ih = (const float*)d_in[16], *c_b_hh = (const float*)d_in[17];
    const float* l_w_ih = (const float*)d_in[18], *l_w_hh = (const float*)d_in[19];
    const float* l_b_ih = (const float*)d_in[20], *l_b_hh = (const float*)d_in[21];

    const int Ls = in_sizes[4] / 128;     // 80000
    const int Cs = in_sizes[5] / 128;     // 160000
    const int nE = in_sizes[2];           // 480000
    const size_t LSZ = (size_t)Ls * 128;
    const size_t CSZ = (size_t)Cs * 128;

    float* lout = (float*)d_out;          // [5][Ls][128]
    float* cout = lout + 5 * LSZ;         // [5][Cs][128]

    // workspace carve
    float* p   = (float*)d_ws;
    float* tmp = p;  p += CSZ;            // MLP hidden
    float* msg = p;  p += CSZ;            // MLP output
    float* l2c = p;  p += CSZ;            // l->c segment sums
    float* c2l = p;  p += LSZ;            // c->l segment sums
    float* gi  = p;  p += (size_t)Cs * 384;
    float* gh  = p;

    // iteration 0 slots = initial embeddings
    hipMemcpyAsync(lout, l_emb0, LSZ * sizeof(float), hipMemcpyDeviceToDevice, stream);
    hipMemcpyAsync(cout, c_emb0, CSZ * sizeof(float), hipMemcpyDeviceToDevice, stream);

    const int scat_blocks = (nE + 7) / 8;

    for (int t = 0; t < 4; ++t) {
        const float* l_old = lout + (size_t)t * LSZ;
        const float* c_old = cout + (size_t)t * CSZ;
        float*       l_new = lout + (size_t)(t + 1) * LSZ;
        float*       c_new = cout + (size_t)(t + 1) * CSZ;

        // literal -> clause messages: MLP(l_old) scattered to clauses
        launch_gemm(0, 128, true,  l_old, nullptr, l2c_w1, l2c_b1, tmp, Ls, 128, stream);
        launch_gemm(0, 128, false, tmp,   nullptr, l2c_w2, l2c_b2, msg, Ls, 128, stream);
        hipMemsetAsync(l2c, 0, CSZ * sizeof(float), stream);
        hipLaunchKernelGGL(scatter_add_kernel, dim3(scat_blocks), dim3(256), 0, stream,
                           msg, l_edge, c_edge, l2c, nE);

        // clause -> literal messages: MLP(c_old) scattered to literals
        launch_gemm(0, 128, true,  c_old, nullptr, c2l_w1, c2l_b1, tmp, Cs, 128, stream);
        launch_gemm(0, 128, false, tmp,   nullptr, c2l_w2, c2l_b2, msg, Cs, 128, stream);
        hipMemsetAsync(c2l, 0, LSZ * sizeof(float), stream);
        hipLaunchKernelGGL(scatter_add_kernel, dim3(scat_blocks), dim3(256), 0, stream,
                           msg, c_edge, l_edge, c2l, nE);

        // clause GRU: c_new = GRU(l2c, c_old)
        launch_gemm(0, 128, false, l2c,   nullptr, c_w_ih, c_b_ih, gi, Cs, 384, stream);
        launch_gemm(0, 128, false, c_old, nullptr, c_w_hh, c_b_hh, gh, Cs, 384, stream);
        hipLaunchKernelGGL(gru_gate_kernel, dim3((unsigned)((CSZ + 255) / 256)), dim3(256),
                           0, stream, gi, gh, c_old, c_new, Cs);

        // literal GRU: l_new = GRU(concat(c2l, swap(l_old)), l_old)
        launch_gemm(1, 256, false, c2l,   l_old,   l_w_ih, l_b_ih, gi, Ls, 384, stream);
        launch_gemm(0, 128, false, l_old, nullptr, l_w_hh, l_b_hh, gh, Ls, 384, stream);
        hipLaunchKernelGGL(gru_gate_kernel, dim3((unsigned)((LSZ + 255) / 256)), dim3(256),
                           0, stream, gi, gh, l_old, l_new, Ls);
    }
}